// PhysicalLoss_34213709480503
// MI455X (gfx1250) — compile-verified
//
#include <hip/hip_runtime.h>
#include <math.h>

// ---------------------------------------------------------------------------
// PhysicalLoss for MI455X (gfx1250, wave32).
//   out = 0.5*mean((pred-targ)^2) + 0.5*mean(||divPK1_neohook(pred)||_2)
// Stage 1: 4x4x16-point tiles; 8x8x20 vec3 halo tile staged into LDS with
//          global_load_async_to_lds_b32 (ASYNCcnt), nested jnp.gradient
//          stencils + neo-Hookean algebra evaluated from LDS, deterministic
//          LDS tree reduction -> 4096 partial (mse,phy) pairs in d_ws.
// Stage 2: single wave32 reduces partials with V_WMMA_F32_16X16X4_F32
//          (B = ones => D[m,n] = sum_k A[m,k] + C[m,n]), writes scalar.
// ---------------------------------------------------------------------------

typedef __attribute__((ext_vector_type(2))) float v2f;
typedef __attribute__((ext_vector_type(8))) float v8f;

#define NPTS   (4 * 64 * 64 * 64)      // 1048576 grid points
#define NBLK   4096                    // tiles: 4 batches * 16*16*4
#define MU_C   1000.0f
#define LMBD_C 5000.0f

// Tile geometry: 4x4x16 points per block (256 threads), halo of 2 each side.
#define TX 4
#define TY 4
#define TZ 16
#define LX 8                           // TX + 4
#define LY 8                           // TY + 4
#define LZ 20                          // TZ + 4
#define NENT (LX * LY * LZ)            // 1280 vec3 entries (15360 B)

__device__ __forceinline__ void loadTile3(const float* __restrict__ t,
                                          int lx, int ly, int lz, float v[3]) {
  const int e = ((lx * LY) + ly) * LZ + lz;
  v[0] = t[e * 3 + 0];
  v[1] = t[e * 3 + 1];
  v[2] = t[e * 3 + 2];
}

// gradU at global point g (jnp.gradient semantics along each length-64 axis),
// reading the staged LDS tile with origin O (halo offset +2).
__device__ __forceinline__ void gradAtL(const float* __restrict__ t,
                                        const int g[3], const int O[3],
                                        float out[3][3]) {
#pragma unroll
  for (int j = 0; j < 3; ++j) {
    int gp[3] = {g[0], g[1], g[2]};
    int gm[3] = {g[0], g[1], g[2]};
    float s;
    if (g[j] == 0)       { gp[j] = 1;        s = 1.0f; }
    else if (g[j] == 63) { gm[j] = 62;       s = 1.0f; }
    else                 { gp[j] = g[j] + 1; gm[j] = g[j] - 1; s = 0.5f; }
    float vp[3], vm[3];
    loadTile3(t, gp[0] - O[0] + 2, gp[1] - O[1] + 2, gp[2] - O[2] + 2, vp);
    loadTile3(t, gm[0] - O[0] + 2, gm[1] - O[1] + 2, gm[2] - O[2] + 2, vm);
#pragma unroll
    for (int i = 0; i < 3; ++i) out[i][j] = s * (vp[i] - vm[i]);
  }
}

__global__ void __launch_bounds__(256)
kmain(const float* __restrict__ pred, const float* __restrict__ targ,
      float* __restrict__ part_mse, float* __restrict__ part_phy) {
  __shared__ float tile_f[NENT * 3];   // 15360 B vec3 tile (+halo 2)
  __shared__ float sm[256];
  __shared__ float sp[256];

  const int tid = threadIdx.x;
  const int blk = blockIdx.x;
  const int tz = blk & 3, ty = (blk >> 2) & 15, tx = (blk >> 6) & 15;
  const int b  = blk >> 10;
  const int O[3] = {tx * TX, ty * TY, tz * TZ};

  // ----- stage tile (+halo, edge-clamped; clamped slots are never read) -----
  // 1280 entries / 256 threads = 5 per thread; 3x b32 async per vec3 entry.
  for (int e = tid; e < NENT; e += 256) {
    const int lz  = e % LZ;
    const int rem = e / LZ;
    const int ly  = rem % LY;
    const int lx  = rem / LY;
    const int gx = min(63, max(0, O[0] + lx - 2));
    const int gy = min(63, max(0, O[1] + ly - 2));
    const int gz = min(63, max(0, O[2] + lz - 2));
    const float* gp = pred + ((((size_t)b * 64 + gx) * 64 + gy) * 64 + gz) * 3;
    // Generic LDS pointer keeps the LDS byte offset in addr[31:0].
    const unsigned la = (unsigned)(size_t)&tile_f[e * 3];
    asm volatile(
        "global_load_async_to_lds_b32 %0, %1, off\n\t"
        "global_load_async_to_lds_b32 %0, %1, off offset:4\n\t"
        "global_load_async_to_lds_b32 %0, %1, off offset:8"
        :: "v"(la), "v"(gp) : "memory");
  }
  asm volatile("s_wait_asynccnt 0x0" ::: "memory");
  __syncthreads();

  // ----- this thread's point -----
  const int sz = tid & 15, sy = (tid >> 4) & 3, sx = tid >> 6;
  const int g[3] = {O[0] + sx, O[1] + sy, O[2] + sz};

  // ----- MSE contribution (pred from LDS tile, targ from global) -----
  float up[3];
  loadTile3(tile_f, sx + 2, sy + 2, sz + 2, up);
  const size_t idx =
      ((((size_t)b * 64 + g[0]) * 64 + g[1]) * 64 + g[2]) * 3;
  float d0 = up[0] - targ[idx + 0];
  float d1 = up[1] - targ[idx + 1];
  float d2 = up[2] - targ[idx + 2];
  float mse = d0 * d0 + d1 * d1 + d2 * d2;

  // ----- gradU at p -----
  float G[3][3];
  gradAtL(tile_f, g, O, G);

  // ----- hessU: nested jnp.gradient of the gradU field -----
  float H[3][3][3];   // H[i][j][k] = D_k gradU[i][j]
#pragma unroll
  for (int k = 0; k < 3; ++k) {
    int cp[3] = {g[0], g[1], g[2]};
    int cm[3] = {g[0], g[1], g[2]};
    if (g[k] == 0) {
      cp[k] = 1;
      float gpv[3][3];
      gradAtL(tile_f, cp, O, gpv);
#pragma unroll
      for (int i = 0; i < 3; ++i)
#pragma unroll
        for (int j = 0; j < 3; ++j) H[i][j][k] = gpv[i][j] - G[i][j];
    } else if (g[k] == 63) {
      cm[k] = 62;
      float gmv[3][3];
      gradAtL(tile_f, cm, O, gmv);
#pragma unroll
      for (int i = 0; i < 3; ++i)
#pragma unroll
        for (int j = 0; j < 3; ++j) H[i][j][k] = G[i][j] - gmv[i][j];
    } else {
      cp[k] = g[k] + 1;
      cm[k] = g[k] - 1;
      float gpv[3][3], gmv[3][3];
      gradAtL(tile_f, cp, O, gpv);
      gradAtL(tile_f, cm, O, gmv);
#pragma unroll
      for (int i = 0; i < 3; ++i)
#pragma unroll
        for (int j = 0; j < 3; ++j) H[i][j][k] = 0.5f * (gpv[i][j] - gmv[i][j]);
    }
  }

  // ----- neo-Hookean residual divPK1 -----
  const int P1[3] = {1, 2, 0}, P2[3] = {2, 0, 1};
  float F[3][3];
#pragma unroll
  for (int i = 0; i < 3; ++i)
#pragma unroll
    for (int j = 0; j < 3; ++j) F[i][j] = G[i][j] + (i == j ? 1.0f : 0.0f);

  float cof[3][3], gcof[3][3][3];
#pragma unroll
  for (int m = 0; m < 3; ++m) {
    const int r1 = P1[m], r2 = P2[m];
#pragma unroll
    for (int i = 0; i < 3; ++i) {
      const int c1 = P1[i], c2 = P2[i];
      cof[m][i] = F[r1][c1] * F[r2][c2] - F[r1][c2] * F[r2][c1];
#pragma unroll
      for (int j = 0; j < 3; ++j)
        gcof[m][i][j] = F[r2][c2] * H[r1][c1][j] + F[r1][c1] * H[r2][c2][j]
                      - F[r2][c1] * H[r1][c2][j] - F[r1][c2] * H[r2][c1][j];
    }
  }

  float J = F[0][0] * cof[0][0] + F[1][0] * cof[1][0] + F[2][0] * cof[2][0] + 1e-8f;
  float invJ = 1.0f / J;

  float gJ[3];
#pragma unroll
  for (int j = 0; j < 3; ++j)
    gJ[j] = cof[0][0] * H[0][0][j] + cof[1][0] * H[1][0][j] + cof[2][0] * H[2][0][j]
          + gcof[0][0][j] * F[0][0] + gcof[1][0][j] * F[1][0] + gcof[2][0][j] * F[2][0];

  float r[3];
#pragma unroll
  for (int m = 0; m < 3; ++m) {
    float t1 = gJ[0] * cof[m][0] + gJ[1] * cof[m][1] + gJ[2] * cof[m][2];
    float t2 = gcof[m][0][0] + gcof[m][1][1] + gcof[m][2][2];
    float dInv = -invJ * invJ * t1 + invJ * t2;
    float dF = H[m][0][0] + H[m][1][1] + H[m][2][2];
    float v = MU_C * (dF - dInv)
            + 0.5f * LMBD_C * (J * J - 1.0f) * dInv
            + LMBD_C * t1;
    r[m] = __builtin_isnan(v) ? 0.0f : v;   // jnp.where(isnan, 0, r)
  }
  float phy = sqrtf(r[0] * r[0] + r[1] * r[1] + r[2] * r[2]);

  // ----- deterministic block reduction -----
  sm[tid] = mse;
  sp[tid] = phy;
  __syncthreads();
#pragma unroll
  for (int off = 128; off > 0; off >>= 1) {
    if (tid < off) {
      sm[tid] += sm[tid + off];
      sp[tid] += sp[tid + off];
    }
    __syncthreads();
  }
  if (tid == 0) {
    part_mse[blockIdx.x] = sm[0];
    part_phy[blockIdx.x] = sp[0];
  }
}

// Stage 2: one wave32 (EXEC all ones, as WMMA requires). Each
// v_wmma_f32_16x16x4_f32 with B=ones folds 64 f32 partials into the resident
// 16x16 f32 accumulator: D[m,n] = sum_k A[m,k] + C[m,n] (columns identical).
__global__ void __launch_bounds__(32)
kreduce(const float* __restrict__ part_mse, const float* __restrict__ part_phy,
        float* __restrict__ out) {
  const int lane = threadIdx.x;            // 0..31, whole wave active
  v2f ones;
  ones[0] = 1.0f;
  ones[1] = 1.0f;
  v8f accM = {};
  v8f accP = {};
#pragma unroll 4
  for (int it = 0; it < NBLK / 64; ++it) { // 64 iterations x 64 values
    const int base = it * 64 + lane * 2;
    v2f a;
    a[0] = part_mse[base + 0];
    a[1] = part_mse[base + 1];
    accM = __builtin_amdgcn_wmma_f32_16x16x4_f32(
        false, a, false, ones, (short)0, accM, false, false);
    v2f p;
    p[0] = part_phy[base + 0];
    p[1] = part_phy[base + 1];
    accP = __builtin_amdgcn_wmma_f32_16x16x4_f32(
        false, p, false, ones, (short)0, accP, false, false);
  }
  // C/D layout: column 0 lives in lane 0 (M=0..7 in VGPRs 0..7) and
  // lane 16 (M=8..15). Sum those to get the full reduction.
  float sM = 0.0f, sP = 0.0f;
#pragma unroll
  for (int q = 0; q < 8; ++q) {
    sM += accM[q];
    sP += accP[q];
  }
  float totM = __shfl(sM, 0, 32) + __shfl(sM, 16, 32);
  float totP = __shfl(sP, 0, 32) + __shfl(sP, 16, 32);
  if (lane == 0) {
    const float invElems = 1.0f / (float)(NPTS * 3);  // mse mean over elements
    const float invPts   = 1.0f / (float)NPTS;        // phy mean over points
    out[0] = 0.5f * (totM * invElems) + 0.5f * (totP * invPts);
  }
}

extern "C" void kernel_launch(void* const* d_in, const int* in_sizes, int n_in,
                              void* d_out, int out_size, void* d_ws, size_t ws_size,
                              hipStream_t stream) {
  (void)in_sizes; (void)n_in; (void)out_size; (void)ws_size;
  const float* pred = (const float*)d_in[0];
  const float* targ = (const float*)d_in[1];
  float* part_mse = (float*)d_ws;          // 4096 floats
  float* part_phy = part_mse + NBLK;       // 4096 floats (32 KB total)

  kmain<<<NBLK, 256, 0, stream>>>(pred, targ, part_mse, part_phy);
  kreduce<<<1, 32, 0, stream>>>(part_mse, part_phy, (float*)d_out);
}